// TSPUpperModel_38946763440478
// MI455X (gfx1250) — compile-verified
//
#include <hip/hip_runtime.h>
#include <math.h>

// Problem constants (match reference)
#define Bsz 128
#define Nsz 2000
#define Usz 1000
#define Dsz 128
#define LOGIT_CLIP 10.0f

typedef float v2f __attribute__((ext_vector_type(2)));
typedef float v8f __attribute__((ext_vector_type(8)));

// D = A(16x4 f32) * B(4x16 f32) + C(16x16 f32)
#define WMMA_F32X4(a, bm, c) \
  __builtin_amdgcn_wmma_f32_16x16x4_f32(false, (a), false, (bm), (short)0, (c), false, false)

// ---------------------------------------------------------------------------
// Kernel A: embed gathered rows, project K/V with fp32 WMMA, accumulate the
// AFT per-dim softmax statistics sumExp[b][d] = sum_u exp(bias_u + K[u][d])
// and sumWV[b][d] = sum_u exp(bias_u + K[u][d]) * V[u][d].
// Grid: (16 row-tiles of 64, B).  Block: 128 threads = 4 waves of 32.
// Each wave owns a 16-row slab; accumulators: 8 n-tiles x (K,V) = 16 x v8f.
// ---------------------------------------------------------------------------
__global__ __launch_bounds__(128)
void kernA(const float* __restrict__ problems, const float* __restrict__ W_embed,
           const float* __restrict__ b_embed, const float* __restrict__ Wk,
           const float* __restrict__ Wv, const float* __restrict__ alpha_attn,
           const float* __restrict__ log_scale, const float* __restrict__ cur_dist,
           const float* __restrict__ ninf, const int* __restrict__ unvis,
           float* __restrict__ sumExp, float* __restrict__ sumWV)
{
  const int b    = blockIdx.y;
  const int u0   = blockIdx.x * 64;
  const int tid  = threadIdx.x;
  const int lane = tid & 31;
  const int wave = tid >> 5;
  const int half = lane >> 4;        // 0: lanes 0-15, 1: lanes 16-31
  const int c    = lane & 15;

  __shared__ float sE[64 * Dsz];     // embedded rows (A-matrix source), 32 KB
  __shared__ float sWk[16 * Dsz];    // Wk K-chunk, 8 KB
  __shared__ float sWv[16 * Dsz];    // Wv K-chunk, 8 KB
  __shared__ float sBias[64];

  // Warm L2 for the weight streams (lowers to global_prefetch_b8).
  __builtin_prefetch(Wk + tid * 16, 0, 1);
  __builtin_prefetch(Wv + tid * 16, 0, 1);

  const float lsA = log_scale[0] * alpha_attn[0];

  // --- stage embeddings: e[u][d] = p0*W0[d] + p1*W1[d] + be[d] -------------
  {
    const int dcol = tid;            // one column per thread
    const float w0 = W_embed[dcol];
    const float w1 = W_embed[Dsz + dcol];
    const float be = b_embed[dcol];
    for (int r = 0; r < 64; ++r) {
      const int u = u0 + r;
      float e = 0.0f;
      if (u < Usz) {
        const int idx = unvis[b * Usz + u];
        const float p0 = problems[(b * Nsz + idx) * 2 + 0];
        const float p1 = problems[(b * Nsz + idx) * 2 + 1];
        e = fmaf(p0, w0, fmaf(p1, w1, be));
      }
      sE[r * Dsz + dcol] = e;
    }
    if (tid < 64) {
      const int u = u0 + tid;
      sBias[tid] = (u < Usz) ? (-lsA * cur_dist[b * Usz + u] + ninf[b * Usz + u])
                             : -1e30f;   // padded rows contribute exp() == 0
    }
  }

  v8f accK[8], accV[8];
  {
    const v8f z = {0.f, 0.f, 0.f, 0.f, 0.f, 0.f, 0.f, 0.f};
    #pragma unroll
    for (int i = 0; i < 8; ++i) { accK[i] = z; accV[i] = z; }
  }

  // --- main WMMA loop: D=128 split into 8 chunks of 16 K-values ------------
  for (int kb = 0; kb < 8; ++kb) {
    __syncthreads();                 // also covers the sE/sBias staging above
    for (int i = tid; i < 16 * Dsz; i += 128) {
      sWk[i] = Wk[kb * 16 * Dsz + i];
      sWv[i] = Wv[kb * 16 * Dsz + i];
    }
    __syncthreads();

    #pragma unroll
    for (int kk = 0; kk < 4; ++kk) {           // 4 K-steps of depth 4
      const int klocal = kk * 4 + 2 * half;    // A/B frag layout (ISA 7.12.2)
      const int kglob  = kb * 16 + klocal;
      const int r      = wave * 16 + c;        // A row for this lane
      v2f a;
      a.x = sE[r * Dsz + kglob];
      a.y = sE[r * Dsz + kglob + 1];
      #pragma unroll
      for (int nt = 0; nt < 8; ++nt) {         // 8 N-tiles of 16
        v2f bk, bv;
        bk.x = sWk[klocal * Dsz + nt * 16 + c];
        bk.y = sWk[(klocal + 1) * Dsz + nt * 16 + c];
        accK[nt] = WMMA_F32X4(a, bk, accK[nt]);
        bv.x = sWv[klocal * Dsz + nt * 16 + c];
        bv.y = sWv[(klocal + 1) * Dsz + nt * 16 + c];
        accV[nt] = WMMA_F32X4(a, bv, accV[nt]);
      }
    }
  }
  __syncthreads();

  // --- AFT epilogue: exp(bias + K) and exp(bias + K)*V, reduced over rows --
  #pragma unroll
  for (int nt = 0; nt < 8; ++nt) {
    float sEx = 0.0f, sWVv = 0.0f;
    #pragma unroll
    for (int i = 0; i < 8; ++i) {
      const int m = wave * 16 + half * 8 + i;  // C/D layout: M = i + 8*half
      const float ex = __expf(sBias[m] + accK[nt][i]);
      sEx  += ex;
      sWVv += ex * accV[nt][i];
    }
    // combine lanes L and L^16 (same N column, complementary M halves)
    sEx  += __shfl_xor(sEx, 16, 32);
    sWVv += __shfl_xor(sWVv, 16, 32);
    if (half == 0) {
      const int n = nt * 16 + c;
      atomicAdd(&sumExp[b * Dsz + n], sEx);
      atomicAdd(&sumWV[b * Dsz + n], sWVv);
    }
  }
}

// ---------------------------------------------------------------------------
// Kernel B: queries + aafm + collapse score dot to 3 scalars per batch.
// q[d] = sum_j e_first[j]*Wq_first[j][d] + e_cur[j]*Wq_last[j][d]
// aafm[d] = sigmoid(q[d]) * sumWV[b][d] / sumExp[b][d]
// c0 = aafm.W_embed[0], c1 = aafm.W_embed[1], c2 = aafm.b_embed
// Grid: B blocks of 128 threads (one output dim per thread).
// ---------------------------------------------------------------------------
__global__ __launch_bounds__(128)
void kernB(const float* __restrict__ problems, const float* __restrict__ W_embed,
           const float* __restrict__ b_embed, const float* __restrict__ Wq_first,
           const float* __restrict__ Wq_last, const int* __restrict__ first_node,
           const int* __restrict__ current_node, const float* __restrict__ sumExp,
           const float* __restrict__ sumWV, float* __restrict__ consts)
{
  const int b = blockIdx.x;
  const int d = threadIdx.x;
  __shared__ float ef[Dsz], ec[Dsz], red[Dsz];

  const int fi = first_node[b];
  const int ci = current_node[b];
  const float w0 = W_embed[d], w1 = W_embed[Dsz + d], be = b_embed[d];
  {
    const float pf0 = problems[(b * Nsz + fi) * 2 + 0];
    const float pf1 = problems[(b * Nsz + fi) * 2 + 1];
    const float pc0 = problems[(b * Nsz + ci) * 2 + 0];
    const float pc1 = problems[(b * Nsz + ci) * 2 + 1];
    ef[d] = fmaf(pf0, w0, fmaf(pf1, w1, be));
    ec[d] = fmaf(pc0, w0, fmaf(pc1, w1, be));
  }
  __syncthreads();

  float q = 0.0f;
  for (int j = 0; j < Dsz; ++j)
    q = fmaf(ef[j], Wq_first[j * Dsz + d], fmaf(ec[j], Wq_last[j * Dsz + d], q));

  const float sig  = 1.0f / (1.0f + __expf(-q));
  const float aafm = sig * sumWV[b * Dsz + d] / sumExp[b * Dsz + d];

  float c012[3];
  const float terms[3] = {aafm * w0, aafm * w1, aafm * be};
  for (int t = 0; t < 3; ++t) {
    red[d] = terms[t];
    __syncthreads();
    for (int off = 64; off > 0; off >>= 1) {
      if (d < off) red[d] += red[d + off];
      __syncthreads();
    }
    c012[t] = red[0];
    __syncthreads();
  }
  if (d == 0) {
    consts[b * 4 + 0] = c012[0];
    consts[b * 4 + 1] = c012[1];
    consts[b * 4 + 2] = c012[2];
  }
}

// ---------------------------------------------------------------------------
// Kernel D: per-batch scores, stable softmax over U, first-occurrence argmax,
// scatter into the zeroed (B,N) output row, plus true_selected/selected_score.
// Grid: B blocks of 256 threads (4 U-slots per thread).
// ---------------------------------------------------------------------------
__global__ __launch_bounds__(256)
void kernD(const float* __restrict__ problems, const int* __restrict__ unvis,
           const float* __restrict__ cur_dist, const float* __restrict__ ninf,
           const float* __restrict__ consts, const float* __restrict__ log_scale,
           const float* __restrict__ alpha_com, float* __restrict__ out)
{
  const int b = blockIdx.x;
  const int tid = threadIdx.x;
  __shared__ float rval[256];
  __shared__ int   ridx[256];
  __shared__ float rsum[256];

  const float c0 = consts[b * 4 + 0];
  const float c1 = consts[b * 4 + 1];
  const float c2 = consts[b * 4 + 2];
  const float lsC = log_scale[0] * alpha_com[0];
  const float invSqrtD = 0.08838834764831845f;   // 1/sqrt(128)

  float* row = out + (size_t)b * Nsz;
  for (int i = tid; i < Nsz; i += 256) row[i] = 0.0f;

  float s[4];
  float m = -1e30f;
  int   mi = 0x7fffffff;
  #pragma unroll
  for (int t = 0; t < 4; ++t) {
    const int u = tid + t * 256;
    float sc = -1e30f;
    if (u < Usz) {
      const int idx = unvis[b * Usz + u];
      const float p0 = problems[(b * Nsz + idx) * 2 + 0];
      const float p1 = problems[(b * Nsz + idx) * 2 + 1];
      const float raw = fmaf(p0, c0, fmaf(p1, c1, c2)) * invSqrtD
                        - lsC * cur_dist[b * Usz + u];
      sc = LOGIT_CLIP * tanhf(raw) + ninf[b * Usz + u];
      if (sc > m) { m = sc; mi = u; }            // per-thread first-max (u ascending)
    }
    s[t] = sc;
  }
  rval[tid] = m; ridx[tid] = mi;
  __syncthreads();
  for (int off = 128; off > 0; off >>= 1) {
    if (tid < off) {
      const float v2 = rval[tid + off];
      const int   i2 = ridx[tid + off];
      if (v2 > rval[tid] || (v2 == rval[tid] && i2 < ridx[tid])) {
        rval[tid] = v2; ridx[tid] = i2;
      }
    }
    __syncthreads();
  }
  const float M  = rval[0];
  const int  sel = ridx[0];
  __syncthreads();

  float ps = 0.0f;
  #pragma unroll
  for (int t = 0; t < 4; ++t) {
    const int u = tid + t * 256;
    if (u < Usz) ps += __expf(s[t] - M);
  }
  rsum[tid] = ps;
  __syncthreads();
  for (int off = 128; off > 0; off >>= 1) {
    if (tid < off) rsum[tid] += rsum[tid + off];
    __syncthreads();
  }
  const float S = rsum[0];

  #pragma unroll
  for (int t = 0; t < 4; ++t) {
    const int u = tid + t * 256;
    if (u < Usz) {
      const int idx = unvis[b * Usz + u];
      row[idx] = __expf(s[t] - M) / S;           // scatter probs
    }
  }
  if (tid == 0) {
    const int trueSel = unvis[b * Usz + sel];
    out[(size_t)Bsz * Nsz + b]        = (float)trueSel;  // true_selected
    out[(size_t)Bsz * Nsz + Bsz + b]  = 1.0f / S;        // selected prob (max => exp(0)/S)
  }
}

// ---------------------------------------------------------------------------
extern "C" void kernel_launch(void* const* d_in, const int* in_sizes, int n_in,
                              void* d_out, int out_size, void* d_ws, size_t ws_size,
                              hipStream_t stream) {
  (void)in_sizes; (void)n_in; (void)out_size; (void)ws_size;
  const float* problems   = (const float*)d_in[0];
  const float* W_embed    = (const float*)d_in[1];
  const float* b_embed    = (const float*)d_in[2];
  const float* Wq_first   = (const float*)d_in[3];
  const float* Wq_last    = (const float*)d_in[4];
  const float* Wk         = (const float*)d_in[5];
  const float* Wv         = (const float*)d_in[6];
  const float* alpha_attn = (const float*)d_in[7];
  const float* alpha_com  = (const float*)d_in[8];
  const float* log_scale  = (const float*)d_in[9];
  const float* cur_dist   = (const float*)d_in[10];
  const float* ninf_mask  = (const float*)d_in[11];
  const int*   current_node = (const int*)d_in[12];
  const int*   first_node   = (const int*)d_in[13];
  const int*   unvisited    = (const int*)d_in[14];

  float* out    = (float*)d_out;
  float* sumExp = (float*)d_ws;                   // B*D
  float* sumWV  = sumExp + Bsz * Dsz;             // B*D
  float* consts = sumWV + Bsz * Dsz;              // B*4

  hipMemsetAsync(d_ws, 0, (size_t)2 * Bsz * Dsz * sizeof(float), stream);

  kernA<<<dim3(16, Bsz), 128, 0, stream>>>(problems, W_embed, b_embed, Wk, Wv,
                                           alpha_attn, log_scale, cur_dist,
                                           ninf_mask, unvisited, sumExp, sumWV);
  kernB<<<Bsz, 128, 0, stream>>>(problems, W_embed, b_embed, Wq_first, Wq_last,
                                 first_node, current_node, sumExp, sumWV, consts);
  kernD<<<Bsz, 256, 0, stream>>>(problems, unvisited, cur_dist, ninf_mask,
                                 consts, log_scale, alpha_com, out);
}